// EKVNonLinearConv_77618648973839
// MI455X (gfx1250) — compile-verified
//
#include <hip/hip_runtime.h>

// EKV nonlinear conv: transcendental-bound (2 exp + 2 log per tap-eval),
// LDS-staged theta + x row slab via gfx1250 async global->LDS copies.

#define BLOCK   256
#define CIN_    16
#define COUT_   64
#define KK_     3
#define HH_     32
#define WW_     32
#define CKK_    (CIN_ * KK_ * KK_)        // 144
#define XROW    (WW_ + 2)                 // 34 (1 pad each side)

#define INV_DENOM   13.333333333333334f   // 1 / (2*1.5*0.025)
#define SHIFT_      1.3333333333333333f   // VD / denom
#define CLIP_LO_    (-50.0f)
#define CLIP_HI_    (20.0f)
#define OUT_SCALE_  0.005625f             // ALPHA * R

#if defined(__AMDGCN__) && \
    __has_builtin(__builtin_amdgcn_global_load_async_to_lds_b32) && \
    __has_builtin(__builtin_amdgcn_global_load_async_to_lds_b128) && \
    __has_builtin(__builtin_amdgcn_s_wait_asynccnt)
#define USE_ASYNC_LDS 1
#else
#define USE_ASYNC_LDS 0
#endif

typedef int v4i __attribute__((ext_vector_type(4)));
#define GAS1 __attribute__((address_space(1)))
#define LAS3 __attribute__((address_space(3)))

__device__ __forceinline__ void async_copy_b32(const float* g, float* l) {
#if USE_ASYNC_LDS
  __builtin_amdgcn_global_load_async_to_lds_b32(
      (GAS1 int*)g, (LAS3 int*)l, 0, 0);
#else
  *l = *g;
#endif
}

__device__ __forceinline__ void async_copy_b128(const float* g, float* l) {
#if USE_ASYNC_LDS
  __builtin_amdgcn_global_load_async_to_lds_b128(
      (GAS1 v4i*)g, (LAS3 v4i*)l, 0, 0);
#else
  *(float4*)l = *(const float4*)g;
#endif
}

__device__ __forceinline__ void async_wait_all() {
#if USE_ASYNC_LDS
  __builtin_amdgcn_s_wait_asynccnt(0);
#endif
}

__global__ __launch_bounds__(BLOCK)
void EKVNonLinearConv_kernel(const float* __restrict__ x,
                             const float* __restrict__ theta,
                             const float* __restrict__ scale,
                             float* __restrict__ out) {
  // theta: 64*144 f32 = 36 KB ; x slab: 16 cin * 3 rows * 34 = 6.4 KB
  __shared__ __align__(16) float theta_s[COUT_ * CKK_];  // 9216
  __shared__ float xs[CIN_ * KK_ * XROW];                // 1632

  const int tid = threadIdx.x;
  const int ho  = blockIdx.x;   // output row
  const int b   = blockIdx.y;   // batch

  // ---- stage theta via async b128: 9216 floats = 2304 float4 = 9 per thread
#pragma unroll
  for (int i = 0; i < 9; ++i) {
    const int v = (tid + i * BLOCK) * 4;
    async_copy_b128(theta + v, theta_s + v);
  }

  // ---- stage 3 x-rows (ho-1, ho, ho+1) for all cin; zero-fill padded rows.
  // 48 segments * 32 interior cols = 1536 = 6 per thread
#pragma unroll
  for (int i = 0; i < 6; ++i) {
    const int e   = tid + i * BLOCK;
    const int seg = e >> 5;          // seg = cin*3 + kh
    const int col = e & 31;
    const int cin = seg / 3;
    const int kh  = seg - 3 * cin;
    const int hr  = ho + kh - 1;
    float* dst = &xs[seg * XROW + 1 + col];
    if ((unsigned)hr < (unsigned)HH_) {
      const float* src = x + ((((b * CIN_) + cin) * HH_ + hr) * WW_ + col);
      async_copy_b32(src, dst);
    } else {
      *dst = 0.0f;                   // top/bottom zero padding
    }
  }
  // left/right column padding (96 cells, disjoint from async targets)
  if (tid < 2 * CIN_ * KK_) {
    const int seg = tid >> 1;
    xs[seg * XROW + ((tid & 1) ? (WW_ + 1) : 0)] = 0.0f;
  }

  async_wait_all();        // s_wait_asynccnt 0
  __syncthreads();

  // ---- clip(theta,1,8) and pre-fold 1/denom so the hot loop does (gi - th)
  // 9216 / 256 = 36 per thread
#pragma unroll
  for (int i = 0; i < 36; ++i) {
    const int idx = tid + i * BLOCK;
    const float v = theta_s[idx];
    theta_s[idx] = fminf(fmaxf(v, 1.0f), 8.0f) * INV_DENOM;
  }
  __syncthreads();

  // wave layout: lane = pixel (theta reads wave-uniform -> LDS broadcast;
  // xs reads stride-1), wave = cout-group of 8
  const int p     = tid & 31;
  const int cbase = (tid >> 5) << 3;
  const float scv = scale[0];

  float acc[8];
#pragma unroll
  for (int c = 0; c < 8; ++c) acc[c] = 0.0f;

#pragma unroll 1
  for (int cin = 0; cin < CIN_; ++cin) {
#pragma unroll
    for (int kh = 0; kh < KK_; ++kh) {
      const float* row = &xs[(cin * KK_ + kh) * XROW + p];
#pragma unroll
      for (int kw = 0; kw < KK_; ++kw) {
        const float gi   = row[kw] * INV_DENOM;            // g/denom
        const int   sidx = (cin * KK_ + kh) * KK_ + kw;
        const float* tcol = &theta_s[cbase * CKK_ + sidx];
#pragma unroll
        for (int c = 0; c < 8; ++c) {
          const float th  = tcol[c * CKK_];                // theta/denom (clipped)
          const float a1  = gi - th;
          const float a1c = fminf(fmaxf(a1,          CLIP_LO_), CLIP_HI_);
          const float a2c = fminf(fmaxf(a1 - SHIFT_, CLIP_LO_), CLIP_HI_);
          const float e1  = __expf(a1c);                   // v_exp_f32
          const float e2  = __expf(a2c);
          const float sp1 = __logf(1.0f + e1);             // v_log_f32
          const float sp2 = __logf(1.0f + e2);
          acc[c] += fmaxf(sp1 * sp1 - sp2 * sp2, 0.0f);    // alpha folded out
        }
      }
    }
  }

  const float kfin = OUT_SCALE_ * scv;                     // ALPHA * R * scale
#pragma unroll
  for (int c = 0; c < 8; ++c) {
    out[(((b * COUT_) + cbase + c) << 10) + (ho << 5) + p] = acc[c] * kfin;
  }
}

extern "C" void kernel_launch(void* const* d_in, const int* in_sizes, int n_in,
                              void* d_out, int out_size, void* d_ws, size_t ws_size,
                              hipStream_t stream) {
  const float* x     = (const float*)d_in[0];
  const float* theta = (const float*)d_in[1];
  const float* scale = (const float*)d_in[2];
  float* out = (float*)d_out;

  const int B = in_sizes[0] / (CIN_ * HH_ * WW_);  // 8
  dim3 grid(HH_, B);                               // (ho, b)
  EKVNonLinearConv_kernel<<<grid, BLOCK, 0, stream>>>(x, theta, scale, out);
}